// TFN_56831007260962
// MI455X (gfx1250) — compile-verified
//
#include <hip/hip_runtime.h>
#include <hip/hip_bf16.h>

typedef __attribute__((ext_vector_type(2))) float v2f;
typedef __attribute__((ext_vector_type(8))) float v8f;

#define BATCH   128
#define HID     101          // 100 hidden + leading 1
#define INDIM   300
#define VT      10201        // 101*101  (K of the big GEMM)
#define VT_PAD  10240        // u row stride (zero padded)
#define NQ      30300        // 300*101  (N of the big GEMM)
#define NQ_PAD  30304        // M2 row stride
#define POST    300
#define OUTD    300
#define KC      64
#define NCHUNK  ((VT + KC - 1) / KC)   // 160

// workspace layout (in floats)
#define OFF_A   0
#define OFF_V   (BATCH * HID)
#define OFF_T   (2 * BATCH * HID)
#define OFF_U   (3 * BATCH * HID)                 // 38784
#define OFF_M2  (OFF_U + BATCH * VT_PAD)          // 1349504

// ---------------------------------------------------------------------------
// Kernel 1: three small encoders -> _a/_v/_t in ws, each [128][101] with
// column 0 == 1.0 (the "ones" concatenation).
// ---------------------------------------------------------------------------
__global__ void tfn_encode(const float* __restrict__ ax, const float* __restrict__ vx,
                           const float* __restrict__ tx,
                           const float* __restrict__ Wa, const float* __restrict__ ba,
                           const float* __restrict__ Wv, const float* __restrict__ bv,
                           const float* __restrict__ Wt, const float* __restrict__ bt,
                           float* __restrict__ ws) {
    int idx = blockIdx.x * blockDim.x + threadIdx.x;
    if (idx >= 3 * BATCH * HID) return;
    int mod = idx / (BATCH * HID);
    int rem = idx - mod * (BATCH * HID);
    int b = rem / HID;
    int j = rem - b * HID;

    const float* x; const float* W; const float* bias; float* o;
    if (mod == 0)      { x = ax; W = Wa; bias = ba; o = ws + OFF_A; }
    else if (mod == 1) { x = vx; W = Wv; bias = bv; o = ws + OFF_V; }
    else               { x = tx; W = Wt; bias = bt; o = ws + OFF_T; }

    float r;
    if (j == 0) {
        r = 1.0f;
    } else {
        float acc = bias[j - 1];
        const float* xr = x + b * INDIM;
        const float* wr = W + (j - 1) * INDIM;
        for (int i = 0; i < INDIM; ++i) acc = fmaf(xr[i], wr[i], acc);
        r = acc;
    }
    o[b * HID + j] = r;
}

// ---------------------------------------------------------------------------
// Kernel 2: u[b, j] = _v[b, j/101] * _t[b, j%101], zero-padded to VT_PAD.
// ---------------------------------------------------------------------------
__global__ void tfn_build_u(float* __restrict__ ws) {
    int idx = blockIdx.x * blockDim.x + threadIdx.x;
    if (idx >= BATCH * VT_PAD) return;
    int b = idx / VT_PAD;
    int j = idx - b * VT_PAD;
    float r = 0.0f;
    if (j < VT) {
        int v = j / 101;
        int t = j - v * 101;
        r = ws[OFF_V + b * HID + v] * ws[OFF_T + b * HID + t];
    }
    ws[OFF_U + idx] = r;
}

// ---------------------------------------------------------------------------
// Kernel 3 (the HBM-streaming GEMM):
//   M2[128, 30300] = u[128, 10201] @ W1_view[30300, 10201]^T
// One WG = 8 wave32 = 256 threads. WG owns 64 q-rows x all 128 batches.
// Wave w: batch rows 16w..16w+15, four 16-wide N-subtiles via
// v_wmma_f32_16x16x4_f32 with LDS-staged tiles.
// ---------------------------------------------------------------------------
__global__ __launch_bounds__(256)
void tfn_main_gemm(const float* __restrict__ W1, float* __restrict__ ws) {
    __shared__ float ldsB[64 * 68];    // [q row][k], stride 68 kills bank conflicts
    __shared__ float ldsA[128 * 68];   // [batch row][k]

    const float* __restrict__ U  = ws + OFF_U;
    float* __restrict__       M2 = ws + OFF_M2;

    const int tid   = threadIdx.x;
    const int wave  = tid >> 5;
    const int lane  = tid & 31;
    const int lh    = lane & 15;              // 0..15
    const int kbase = (lane >> 4) << 1;       // 0 or 2 (K half per ISA layout)
    const int qbase = blockIdx.x * 64;

    v8f acc0 = {}; v8f acc1 = {}; v8f acc2 = {}; v8f acc3 = {};

    for (int c = 0; c < NCHUNK; ++c) {
        const int k0 = c * KC;

        // ---- stage W1 chunk (64 rows x 64 k) : coalesced dword stream ----
        for (int idx = tid; idx < 64 * 64; idx += 256) {
            int row = idx >> 6, col = idx & 63;
            int q = qbase + row, k = k0 + col;
            float v = 0.0f;
            if (q < NQ && k < VT) v = W1[q * VT + k];
            ldsB[row * 68 + col] = v;
        }
        // ---- stage u chunk (128 rows x 64 k) : float4 (padded, aligned) ----
        for (int idx = tid; idx < 128 * 16; idx += 256) {
            int row = idx >> 4, c4 = (idx & 15) << 2;
            const float4 src = *reinterpret_cast<const float4*>(U + row * VT_PAD + k0 + c4);
            *reinterpret_cast<float4*>(&ldsA[row * 68 + c4]) = src;
        }
        // ---- prefetch next W1 chunk (global_prefetch_b8) ----
        {
            int nk = k0 + KC;
            if (nk < VT) {
                int prow = tid & 63, pcol = (tid >> 6) << 4;
                int q = qbase + prow;
                if (q < NQ) __builtin_prefetch(W1 + q * VT + nk + pcol, 0, 1);
            }
        }
        __syncthreads();

        // ---- compute: 16 K-steps x 4 N-subtiles of v_wmma_f32_16x16x4_f32 ----
        const float* aRow = &ldsA[(wave * 16 + lh) * 68];
#pragma unroll 4
        for (int kk = 0; kk < KC / 4; ++kk) {
            const int kb = kk * 4 + kbase;
            v2f a = *reinterpret_cast<const v2f*>(aRow + kb);
            v2f b0 = *reinterpret_cast<const v2f*>(&ldsB[(0 * 16 + lh) * 68 + kb]);
            v2f b1 = *reinterpret_cast<const v2f*>(&ldsB[(1 * 16 + lh) * 68 + kb]);
            v2f b2 = *reinterpret_cast<const v2f*>(&ldsB[(2 * 16 + lh) * 68 + kb]);
            v2f b3 = *reinterpret_cast<const v2f*>(&ldsB[(3 * 16 + lh) * 68 + kb]);
            acc0 = __builtin_amdgcn_wmma_f32_16x16x4_f32(false, a, false, b0, (short)0, acc0, false, false);
            acc1 = __builtin_amdgcn_wmma_f32_16x16x4_f32(false, a, false, b1, (short)0, acc1, false, false);
            acc2 = __builtin_amdgcn_wmma_f32_16x16x4_f32(false, a, false, b2, (short)0, acc2, false, false);
            acc3 = __builtin_amdgcn_wmma_f32_16x16x4_f32(false, a, false, b3, (short)0, acc3, false, false);
        }
        __syncthreads();
    }

    // ---- write back: C/D layout -> row r in VGPR r, +8 for upper lane half ----
    const int mofs = (lane >> 4) << 3;   // 0 or 8
#pragma unroll
    for (int r = 0; r < 8; ++r) {
        const int m = wave * 16 + r + mofs;
        float* dst = M2 + m * NQ_PAD;
        int q0 = qbase + 0 * 16 + lh;
        int q1 = qbase + 1 * 16 + lh;
        int q2 = qbase + 2 * 16 + lh;
        int q3 = qbase + 3 * 16 + lh;
        if (q0 < NQ) dst[q0] = acc0[r];
        if (q1 < NQ) dst[q1] = acc1[r];
        if (q2 < NQ) dst[q2] = acc2[r];
        if (q3 < NQ) dst[q3] = acc3[r];
    }
}

// ---------------------------------------------------------------------------
// Kernel 4: contract the 'a' factor + bias + relu, then the 300x300 head.
// One block per batch element.
// ---------------------------------------------------------------------------
__global__ __launch_bounds__(320)
void tfn_final(const float* __restrict__ b1, const float* __restrict__ W2,
               const float* __restrict__ b2, const float* __restrict__ ws,
               float* __restrict__ out) {
    __shared__ float hsh[POST];
    const int b = blockIdx.x;
    const int t = threadIdx.x;

    if (t < POST) {
        const float* A  = ws + OFF_A + b * HID;
        const float* M2 = ws + OFF_M2 + b * NQ_PAD + t * HID;
        float acc = b1[t];
        for (int a = 0; a < HID; ++a) acc = fmaf(A[a], M2[a], acc);
        hsh[t] = fmaxf(acc, 0.0f);
    }
    __syncthreads();
    if (t < OUTD) {
        const float* wr = W2 + t * POST;
        float acc = b2[t];
        for (int p = 0; p < POST; ++p) acc = fmaf(hsh[p], wr[p], acc);
        out[b * OUTD + t] = fmaxf(acc, 0.0f);
    }
}

// ---------------------------------------------------------------------------
extern "C" void kernel_launch(void* const* d_in, const int* in_sizes, int n_in,
                              void* d_out, int out_size, void* d_ws, size_t ws_size,
                              hipStream_t stream) {
    const float* audio_x = (const float*)d_in[0];
    const float* video_x = (const float*)d_in[1];
    const float* text_x  = (const float*)d_in[2];
    const float* Wa = (const float*)d_in[3];
    const float* ba = (const float*)d_in[4];
    const float* Wv = (const float*)d_in[5];
    const float* bv = (const float*)d_in[6];
    const float* Wt = (const float*)d_in[7];
    const float* bt = (const float*)d_in[8];
    const float* W1 = (const float*)d_in[9];
    const float* b1 = (const float*)d_in[10];
    const float* W2 = (const float*)d_in[11];
    const float* b2 = (const float*)d_in[12];
    float* out = (float*)d_out;
    float* ws  = (float*)d_ws;

    // 1) encoders (+ leading ones)
    {
        int total = 3 * BATCH * HID;
        tfn_encode<<<(total + 255) / 256, 256, 0, stream>>>(
            audio_x, video_x, text_x, Wa, ba, Wv, bv, Wt, bt, ws);
    }
    // 2) u = _v (x) _t, zero padded
    {
        int total = BATCH * VT_PAD;
        tfn_build_u<<<(total + 255) / 256, 256, 0, stream>>>(ws);
    }
    // 3) the 79-GFLOP / 1.24-GB streaming GEMM (WMMA f32)
    {
        int ngroups = (NQ + 63) / 64;   // 474
        tfn_main_gemm<<<ngroups, 256, 0, stream>>>(W1, ws);
    }
    // 4) fold the 'a' dimension + relu + output head
    tfn_final<<<BATCH, 320, 0, stream>>>(b1, W2, b2, ws, out);
}